// DropoutDPP_38732015075997
// MI455X (gfx1250) — compile-verified
//
#include <hip/hip_runtime.h>
#include <stdint.h>

// ---------------------------------------------------------------------------
// DropoutDPP eval path for MI455X (gfx1250).
//  Kernel 1: serial RNG phase (JAX threefry2x32 reimplementation), 1 block.
//  Kernel 2: bandwidth-bound scale of x[4,4096,4096] by mask[h]/i.
//            Uses the Tensor Data Mover to stage the 16KB scale vector into
//            LDS per workgroup (tensor_load_to_lds + s_wait_tensorcnt), then
//            streams x with b128 loads/stores + global_prefetch.
// ---------------------------------------------------------------------------

typedef __attribute__((ext_vector_type(4))) unsigned int u32x4;
typedef __attribute__((ext_vector_type(8))) int i32x8;
typedef __attribute__((ext_vector_type(4))) int i32x4;

#define HIDDEN      4096
#define HALF_H      2048
#define DPP_MAX_N   100
#define DPP_P       0.9f
#define DPP_MAXFRAC 0.4f
#define ROWS_PER_BLOCK 8

// ----------------------------- Threefry-2x32 -------------------------------
__device__ __forceinline__ void threefry2x32(uint32_t k0, uint32_t k1,
                                             uint32_t x0, uint32_t x1,
                                             uint32_t& o0, uint32_t& o1) {
  const uint32_t ks2 = k0 ^ k1 ^ 0x1BD11BDAu;
  x0 += k0; x1 += k1;
#define TF_R(r) { x0 += x1; x1 = (x1 << (r)) | (x1 >> (32 - (r))); x1 ^= x0; }
#define TF_4A  TF_R(13) TF_R(15) TF_R(26) TF_R(6)
#define TF_4B  TF_R(17) TF_R(29) TF_R(16) TF_R(24)
  TF_4A; x0 += k1;  x1 += ks2 + 1u;
  TF_4B; x0 += ks2; x1 += k0  + 2u;
  TF_4A; x0 += k0;  x1 += k1  + 3u;
  TF_4B; x0 += k1;  x1 += ks2 + 4u;
  TF_4A; x0 += ks2; x1 += k0  + 5u;
#undef TF_4B
#undef TF_4A
#undef TF_R
  o0 = x0; o1 = x1;
}

// JAX uniform(f32): u = bitcast<f32>((bits >> 9) | 0x3f800000) - 1.0  in [0,1)
__device__ __forceinline__ float bits_to_mask(uint32_t bits) {
  const float u = __uint_as_float((bits >> 9) | 0x3f800000u) - 1.0f;
  return (u >= DPP_P) ? 1.0f : 0.0f;
}

// jax.random.split(key): counts iota(4) -> lanes (0,2) and (1,3);
// new_key = (o0_lane0, o0_lane1), sub_key = (o1_lane0, o1_lane1)
__device__ __forceinline__ void jax_split(uint32_t& key_hi, uint32_t& key_lo,
                                          uint32_t& sub_hi, uint32_t& sub_lo) {
  uint32_t a0, b0, a1, b1;
  threefry2x32(key_hi, key_lo, 0u, 2u, a0, b0);
  threefry2x32(key_hi, key_lo, 1u, 3u, a1, b1);
  key_hi = a0; key_lo = a1;
  sub_hi = b0; sub_lo = b1;
}

// ------------------------ Kernel 1: mask accumulation ----------------------
// Single block of 256 threads; each thread owns threefry lanes t*8..t*8+7,
// i.e. neurons [t*8, t*8+8) and [2048 + t*8, 2048 + t*8+8).
__global__ __launch_bounds__(256) void dpp_mask_kernel(float* __restrict__ mask_out) {
  const int t = threadIdx.x;
  const uint32_t lane0 = (uint32_t)(t * 8);
  __shared__ int s_cnt;

  // key = jax.random.key(42) -> data (hi, lo) = (0, 42); then key, k0 = split(key)
  uint32_t key_hi = 0u, key_lo = 42u;
  uint32_t sk_hi, sk_lo;
  jax_split(key_hi, key_lo, sk_hi, sk_lo);

  float acc[16];
#pragma unroll
  for (int u = 0; u < 8; ++u) {
    uint32_t o0, o1;
    threefry2x32(sk_hi, sk_lo, lane0 + (uint32_t)u, lane0 + (uint32_t)u + HALF_H, o0, o1);
    acc[u]     = bits_to_mask(o0);   // neuron lane0+u
    acc[8 + u] = bits_to_mask(o1);   // neuron 2048+lane0+u
  }

  int i = 1;
  for (;;) {
    // frac_nonzero(sum_mask): exact — cnt is an integer <= 4096 = 2^12
    int local = 0;
#pragma unroll
    for (int u = 0; u < 16; ++u) local += (acc[u] != 0.0f) ? 1 : 0;
    if (t == 0) s_cnt = 0;
    __syncthreads();
    atomicAdd(&s_cnt, local);
    __syncthreads();
    const float frac = (float)s_cnt * (1.0f / 4096.0f);
    __syncthreads();   // everyone read s_cnt before t0 can reset it next iter

    if (!(i < DPP_MAX_N && frac < DPP_MAXFRAC)) break;

    jax_split(key_hi, key_lo, sk_hi, sk_lo);
#pragma unroll
    for (int u = 0; u < 8; ++u) {
      uint32_t o0, o1;
      threefry2x32(sk_hi, sk_lo, lane0 + (uint32_t)u, lane0 + (uint32_t)u + HALF_H, o0, o1);
      acc[u]     += bits_to_mask(o0);
      acc[8 + u] += bits_to_mask(o1);
    }
    ++i;
  }

  const float fi = (float)i;
#pragma unroll
  for (int u = 0; u < 8; ++u) {
    mask_out[lane0 + u]          = acc[u]     / fi;
    mask_out[HALF_H + lane0 + u] = acc[8 + u] / fi;
  }
}

// ------------------------ Kernel 2: streaming scale ------------------------
// Each block: TDM-load the 16KB scale vector into LDS once, then process
// ROWS_PER_BLOCK full hidden rows (4096 floats each) with float4 traffic.
__global__ __launch_bounds__(256) void dpp_scale_kernel(const float* __restrict__ x,
                                                        const float* __restrict__ mask,
                                                        float* __restrict__ out,
                                                        int rows) {
  __shared__ float s_mask[HIDDEN];
  const int t = threadIdx.x;

  if (t < 32) {  // wave 0 only: TDM is a per-wave DMA op (EXEC ignored)
    const uint64_t g   = (uint64_t)(uintptr_t)mask;
    const uint32_t lds = (uint32_t)(uintptr_t)(void*)&s_mask[0]; // low 32b = LDS offset

    // D# group 0: count=1(valid), lds_addr, global_addr[56:0], type=2
    u32x4 g0;
    g0.x = 1u;
    g0.y = lds;
    g0.z = (uint32_t)g;
    g0.w = ((uint32_t)((g >> 32) & 0x01FFFFFFull)) | 0x80000000u;

    // D# group 1: data_size=4B (bits17:16=2); tensor_dim0=4096 (bits79:48);
    // tensor_dim1=1 (bits111:80); tile_dim0=4096 (bits127:112);
    // tile_dim1=1 (bits143:128); tensor_dim0_stride=4096 (bits207:160)
    i32x8 g1 = { (int)0x00020000, (int)0x10000000, (int)0x00010000,
                 (int)0x10000000, 1, 4096, 0, 0 };
    i32x4 g2 = { 0, 0, 0, 0 };
    i32x4 g3 = { 0, 0, 0, 0 };
    i32x8 g4 = { 0, 0, 0, 0, 0, 0, 0, 0 };   // 6-arg toolchain: extra group, zero-filled

    __builtin_amdgcn_tensor_load_to_lds(g0, g1, g2, g3, g4, 0);   // TENSORcnt++
    __builtin_amdgcn_s_wait_tensorcnt(0);                         // s_wait_tensorcnt 0
  }
  __syncthreads();

  const float4* m4 = (const float4*)s_mask;
  const size_t block_base = (size_t)blockIdx.x * ROWS_PER_BLOCK * HIDDEN;

#pragma unroll
  for (int r = 0; r < ROWS_PER_BLOCK; ++r) {
    const int row = blockIdx.x * ROWS_PER_BLOCK + r;
    if (row >= rows) break;
    const size_t base = block_base + (size_t)r * HIDDEN;
    const float4* xr = (const float4*)(x + base);
    float4*       yr = (float4*)(out + base);

    if (r + 1 < ROWS_PER_BLOCK)  // speculative prefetch of next row's slice
      __builtin_prefetch(x + base + HIDDEN + (size_t)t * 4, 0, 0);

#pragma unroll
    for (int j = 0; j < 4; ++j) {
      const int c = t + j * 256;          // float4 index within the row
      const float4 v = xr[c];             // global_load_b128 (coalesced)
      const float4 m = m4[c];             // ds_load_b128
      float4 w;
      w.x = v.x * m.x; w.y = v.y * m.y; w.z = v.z * m.z; w.w = v.w * m.w;
      yr[c] = w;                          // global_store_b128
    }
  }
}

// ------------------------------- Launcher ----------------------------------
extern "C" void kernel_launch(void* const* d_in, const int* in_sizes, int n_in,
                              void* d_out, int out_size, void* d_ws, size_t ws_size,
                              hipStream_t stream) {
  const float* x   = (const float*)d_in[0];
  float*       out = (float*)d_out;
  float*       msk = (float*)d_ws;            // 4096 floats (16 KB) of scratch

  const int total = in_sizes[0];              // 4*4096*4096
  const int rows  = total / HIDDEN;           // 16384
  const int grid  = (rows + ROWS_PER_BLOCK - 1) / ROWS_PER_BLOCK;

  dpp_mask_kernel<<<1, 256, 0, stream>>>(msk);
  dpp_scale_kernel<<<grid, 256, 0, stream>>>(x, msk, out, rows);
}